// _LoRA_qkv_4681514353405
// MI455X (gfx1250) — compile-verified
//
#include <hip/hip_runtime.h>

typedef float v2f __attribute__((ext_vector_type(2)));
typedef float v8f __attribute__((ext_vector_type(8)));
typedef int   v4i_vs __attribute__((vector_size(16)));   // matches builtin's pointee type

#define DIM     1024
#define N3      3072
#define RANK    16
#define BM      128
#define BN      64
#define BK      32
#define LSTRIDE 36      // 32 + 4 pad: distinct banks for 16 fragment rows, keeps 16B alignment
#define P_COLS  48

#define WMMA_F32(a, b, c) \
    __builtin_amdgcn_wmma_f32_16x16x4_f32(false, (a), false, (b), (short)0, (c), false, false)

// -------- gfx1250 async global->LDS path (ASYNCcnt), with sync fallback ----
#if defined(__AMDGCN__) && __has_builtin(__builtin_amdgcn_global_load_async_to_lds_b128)
#define HAS_ASYNC_LDS 1
#define ASYNC_CP_B128(gsrc, ldst)                                              \
    __builtin_amdgcn_global_load_async_to_lds_b128(                            \
        (__attribute__((address_space(1))) v4i_vs*)(const void*)(gsrc),        \
        (__attribute__((address_space(3))) v4i_vs*)(void*)(ldst), 0, 0)
#if __has_builtin(__builtin_amdgcn_s_wait_asynccnt)
#define WAIT_ASYNC(n) __builtin_amdgcn_s_wait_asynccnt(n)
#else
#define WAIT_ASYNC(n) asm volatile("s_wait_asynccnt %0" ::"i"(n) : "memory")
#endif
#else
#define HAS_ASYNC_LDS 0
#endif

// ---------------------------------------------------------------------------
// Kernel 1: P[M, 48] = x @ [A_q; A_k; A_v]^T   (rank-16 projections, fp32 WMMA)
// ---------------------------------------------------------------------------
__global__ __launch_bounds__(256) void lora_p_kernel(
    const float* __restrict__ x,
    const float* __restrict__ Aq, const float* __restrict__ Ak,
    const float* __restrict__ Av,
    float* __restrict__ P)
{
    __shared__ float xs[BM * LSTRIDE];
    __shared__ float as[P_COLS * LSTRIDE];

    const int t      = threadIdx.x;
    const int lane   = t & 31;
    const int w      = t >> 5;               // 8 waves; wave w owns rows [w*16, w*16+16)
    const int blockM = blockIdx.x * BM;

    v8f c0 = {}, c1 = {}, c2 = {};

    const int n15   = lane & 15;
    const int khalf = (lane >> 4) << 1;      // ISA A/B layout: lanes 16-31 hold K+2

    for (int kb = 0; kb < DIM; kb += BK) {
        #pragma unroll
        for (int i = 0; i < 4; ++i) {        // 128x32 x tile
            int lin = t + i * 256;
            int row = lin >> 3, c4 = (lin & 7) << 2;
            *(float4*)(&xs[row * LSTRIDE + c4]) =
                *(const float4*)(x + (size_t)(blockM + row) * DIM + kb + c4);
        }
        #pragma unroll
        for (int i = 0; i < 2; ++i) {        // 48x32 adapter tile (A_q / A_k / A_v)
            int lin = t + i * 256;
            if (lin < 384) {
                int row = lin >> 3, c4 = (lin & 7) << 2;
                const float* Ap = (row < 16) ? (Aq + (size_t)row * DIM)
                                : (row < 32) ? (Ak + (size_t)(row - 16) * DIM)
                                             : (Av + (size_t)(row - 32) * DIM);
                *(float4*)(&as[row * LSTRIDE + c4]) = *(const float4*)(Ap + kb + c4);
            }
        }
        __syncthreads();

        const float* xrow = &xs[(w * 16 + n15) * LSTRIDE + khalf];
        #pragma unroll
        for (int kk = 0; kk < BK; kk += 4) {
            v2f a  = *(const v2f*)(xrow + kk);
            v2f b0 = *(const v2f*)(&as[(0 * 16 + n15) * LSTRIDE + kk + khalf]);
            v2f b1 = *(const v2f*)(&as[(1 * 16 + n15) * LSTRIDE + kk + khalf]);
            v2f b2 = *(const v2f*)(&as[(2 * 16 + n15) * LSTRIDE + kk + khalf]);
            c0 = WMMA_F32(a, b0, c0);
            c1 = WMMA_F32(a, b1, c1);
            c2 = WMMA_F32(a, b2, c2);
        }
        __syncthreads();
    }

    // C/D layout: VGPR v -> M = v + 8*(lane>=16), N = lane&15
    const int mrow = w * 16 + ((lane >> 4) << 3);
    #pragma unroll
    for (int v = 0; v < 8; ++v) {
        size_t m = (size_t)(blockM + mrow + v);
        P[m * P_COLS +  0 + n15] = c0[v];
        P[m * P_COLS + 16 + n15] = c1[v];
        P[m * P_COLS + 32 + n15] = c2[v];
    }
}

// ---------------------------------------------------------------------------
// Kernel 2: out = x @ W_qkv^T + b + alpha * (P_g @ B_g^T)   (fused epilogue)
// Double-buffered async global->LDS staging overlapped with WMMA compute.
// ---------------------------------------------------------------------------
__global__ __launch_bounds__(256) void lora_qkv_gemm(
    const float* __restrict__ x,  const float* __restrict__ W,
    const float* __restrict__ bias,
    const float* __restrict__ Bq, const float* __restrict__ Bk,
    const float* __restrict__ Bv,
    const float* __restrict__ P,  const float* __restrict__ alpha_p,
    float* __restrict__ out)
{
    __shared__ float xs[2][BM * LSTRIDE];
    __shared__ float wsm[2][BN * LSTRIDE];
    __shared__ float pt[BM * P_COLS];

    const int t      = threadIdx.x;
    const int lane   = t & 31;
    const int w      = t >> 5;
    const int blockN = blockIdx.x * BN;
    const int blockM = blockIdx.y * BM;
    const int waveM  = (w >> 1) * 32;        // 4x2 wave grid, 32x32 per wave
    const int waveN  = (w & 1) * 32;

    // stage 128x48 P tile once per block
    #pragma unroll
    for (int i = 0; i < 6; ++i) {
        int lin = t + i * 256;               // 1536 float4 total
        int row = lin / 12, c4 = (lin % 12) << 2;
        *(float4*)(&pt[row * P_COLS + c4]) =
            *(const float4*)(P + (size_t)(blockM + row) * P_COLS + c4);
    }

    v8f c[2][2] = {};

    const int n15   = lane & 15;
    const int khalf = (lane >> 4) << 1;
    const int NT    = DIM / BK;              // 32 K-tiles

#if HAS_ASYNC_LDS
    // ---- async double-buffered pipeline: 6 b128 copies per thread per tile
    {
        #pragma unroll
        for (int i = 0; i < 4; ++i) {
            int lin = t + i * 256;
            int row = lin >> 3, c4 = (lin & 7) << 2;
            ASYNC_CP_B128(x + (size_t)(blockM + row) * DIM + c4,
                          &xs[0][row * LSTRIDE + c4]);
        }
        #pragma unroll
        for (int i = 0; i < 2; ++i) {
            int lin = t + i * 256;
            int row = lin >> 3, c4 = (lin & 7) << 2;
            ASYNC_CP_B128(W + (size_t)(blockN + row) * DIM + c4,
                          &wsm[0][row * LSTRIDE + c4]);
        }
    }
    for (int kt = 0; kt < NT; ++kt) {
        const int cur = kt & 1;
        if (kt + 1 < NT) {                   // prefetch next tile into other buffer
            const int kb = (kt + 1) * BK, nxt = cur ^ 1;
            #pragma unroll
            for (int i = 0; i < 4; ++i) {
                int lin = t + i * 256;
                int row = lin >> 3, c4 = (lin & 7) << 2;
                ASYNC_CP_B128(x + (size_t)(blockM + row) * DIM + kb + c4,
                              &xs[nxt][row * LSTRIDE + c4]);
            }
            #pragma unroll
            for (int i = 0; i < 2; ++i) {
                int lin = t + i * 256;
                int row = lin >> 3, c4 = (lin & 7) << 2;
                ASYNC_CP_B128(W + (size_t)(blockN + row) * DIM + kb + c4,
                              &wsm[nxt][row * LSTRIDE + c4]);
            }
            WAIT_ASYNC(6);                   // tile kt's 6 copies retired (in-order)
        } else {
            WAIT_ASYNC(0);
        }
        __syncthreads();
        #pragma unroll
        for (int kk = 0; kk < BK; kk += 4) {
            v2f a0 = *(const v2f*)(&xs[cur][(waveM      + n15) * LSTRIDE + kk + khalf]);
            v2f a1 = *(const v2f*)(&xs[cur][(waveM + 16 + n15) * LSTRIDE + kk + khalf]);
            v2f b0 = *(const v2f*)(&wsm[cur][(waveN      + n15) * LSTRIDE + kk + khalf]);
            v2f b1 = *(const v2f*)(&wsm[cur][(waveN + 16 + n15) * LSTRIDE + kk + khalf]);
            c[0][0] = WMMA_F32(a0, b0, c[0][0]);
            c[0][1] = WMMA_F32(a0, b1, c[0][1]);
            c[1][0] = WMMA_F32(a1, b0, c[1][0]);
            c[1][1] = WMMA_F32(a1, b1, c[1][1]);
        }
        __syncthreads();                     // protects buffer reuse by next prefetch
    }
#else
    // ---- fallback: synchronous single-buffer staging
    for (int kt = 0; kt < NT; ++kt) {
        const int kb = kt * BK;
        #pragma unroll
        for (int i = 0; i < 4; ++i) {
            int lin = t + i * 256;
            int row = lin >> 3, c4 = (lin & 7) << 2;
            *(float4*)(&xs[0][row * LSTRIDE + c4]) =
                *(const float4*)(x + (size_t)(blockM + row) * DIM + kb + c4);
        }
        #pragma unroll
        for (int i = 0; i < 2; ++i) {
            int lin = t + i * 256;
            int row = lin >> 3, c4 = (lin & 7) << 2;
            *(float4*)(&wsm[0][row * LSTRIDE + c4]) =
                *(const float4*)(W + (size_t)(blockN + row) * DIM + kb + c4);
        }
        __syncthreads();
        #pragma unroll
        for (int kk = 0; kk < BK; kk += 4) {
            v2f a0 = *(const v2f*)(&xs[0][(waveM      + n15) * LSTRIDE + kk + khalf]);
            v2f a1 = *(const v2f*)(&xs[0][(waveM + 16 + n15) * LSTRIDE + kk + khalf]);
            v2f b0 = *(const v2f*)(&wsm[0][(waveN      + n15) * LSTRIDE + kk + khalf]);
            v2f b1 = *(const v2f*)(&wsm[0][(waveN + 16 + n15) * LSTRIDE + kk + khalf]);
            c[0][0] = WMMA_F32(a0, b0, c[0][0]);
            c[0][1] = WMMA_F32(a0, b1, c[0][1]);
            c[1][0] = WMMA_F32(a1, b0, c[1][0]);
            c[1][1] = WMMA_F32(a1, b1, c[1][1]);
        }
        __syncthreads();
    }
#endif

    const float alpha = *alpha_p;

    #pragma unroll
    for (int mi = 0; mi < 2; ++mi) {
        #pragma unroll
        for (int ni = 0; ni < 2; ++ni) {
            const int nloc = waveN + ni * 16 + n15;
            const int ng   = blockN + nloc;          // 0..3071
            const int g    = ng >> 10;               // 0=q, 1=k, 2=v
            const int n1   = ng & 1023;
            const float* Bg = (g == 0) ? Bq : (g == 1) ? Bk : Bv;
            float4 br0 = *(const float4*)(Bg + (size_t)n1 * RANK +  0);
            float4 br1 = *(const float4*)(Bg + (size_t)n1 * RANK +  4);
            float4 br2 = *(const float4*)(Bg + (size_t)n1 * RANK +  8);
            float4 br3 = *(const float4*)(Bg + (size_t)n1 * RANK + 12);
            const float bb = bias[ng];

            #pragma unroll
            for (int v = 0; v < 8; ++v) {
                const int mloc = waveM + mi * 16 + ((lane >> 4) << 3) + v;
                const float* pr = &pt[mloc * P_COLS + g * 16];
                float d = br0.x * pr[0]  + br0.y * pr[1]  + br0.z * pr[2]  + br0.w * pr[3]
                        + br1.x * pr[4]  + br1.y * pr[5]  + br1.z * pr[6]  + br1.w * pr[7]
                        + br2.x * pr[8]  + br2.y * pr[9]  + br2.z * pr[10] + br2.w * pr[11]
                        + br3.x * pr[12] + br3.y * pr[13] + br3.z * pr[14] + br3.w * pr[15];
                out[(size_t)(blockM + mloc) * N3 + ng] = c[mi][ni][v] + bb + alpha * d;
            }
        }
    }
}

// ---------------------------------------------------------------------------
extern "C" void kernel_launch(void* const* d_in, const int* in_sizes, int n_in,
                              void* d_out, int out_size, void* d_ws, size_t ws_size,
                              hipStream_t stream) {
    const float* x     = (const float*)d_in[0];
    const float* Wqkv  = (const float*)d_in[1];
    const float* bqkv  = (const float*)d_in[2];
    const float* Aq    = (const float*)d_in[3];
    const float* Bq    = (const float*)d_in[4];
    const float* Ak    = (const float*)d_in[5];
    const float* Bk    = (const float*)d_in[6];
    const float* Av    = (const float*)d_in[7];
    const float* Bv    = (const float*)d_in[8];
    const float* alpha = (const float*)d_in[9];

    float* P   = (float*)d_ws;         // 8192 x 48 fp32 = 1.5 MB scratch
    float* out = (float*)d_out;

    const int M = 2 * 4096;

    lora_p_kernel<<<M / BM, 256, 0, stream>>>(x, Aq, Ak, Av, P);

    dim3 grid2(N3 / BN, M / BM);       // 48 x 64 blocks
    lora_qkv_gemm<<<grid2, 256, 0, stream>>>(x, Wqkv, bqkv, Bq, Bk, Bv, P, alpha, out);
}